// NAM_23725399343118
// MI455X (gfx1250) — compile-verified
//
#include <hip/hip_runtime.h>

// Problem constants (match reference)
#define B_  2048
#define F_  2048      // K*L = 8*256 features
#define H1_ 18
#define H2_ 12
#define EPS_ 1e-8f

typedef __attribute__((ext_vector_type(2))) float v2f;
typedef __attribute__((ext_vector_type(8))) float v8f;

// One wave per feature. Transposed-GEMM formulation:
//   D = W2^T (16x4 chunks, A operand) x h1^T (4x16, B operand)
// so D[j, m] = z[m, j] lands with hidden index j in the register index and
// samples m across lanes -> layer-3 reduction is register-local + one shuffle,
// and the output store is a full-wave coalesced-in-lane store.
__global__ void __launch_bounds__(256) nam_mlp_wmma_kernel(
    const float* __restrict__ x,
    const float* __restrict__ W1, const float* __restrict__ b1,
    const float* __restrict__ W2, const float* __restrict__ b2,
    const float* __restrict__ W3, const float* __restrict__ b3,
    float* __restrict__ out)
{
    const int lane = threadIdx.x & 31;
    const int wv   = threadIdx.x >> 5;
    const int f    = blockIdx.x * 8 + wv;   // feature id, grid covers exactly F_

    const int g = lane >> 4;    // lane half: selects K pair / D row group
    const int n = lane & 15;    // A row (=j) / B,D column (=sample-in-tile)

    // ---- Pass 1: batch mean / std for this feature ----
    float s = 0.f, q = 0.f;
    #pragma unroll 8
    for (int b = lane; b < B_; b += 32) {
        float v = x[(size_t)b * F_ + f];
        s += v;
        q += v * v;
    }
    #pragma unroll
    for (int mask = 16; mask >= 1; mask >>= 1) {
        s += __shfl_xor(s, mask, 32);
        q += __shfl_xor(q, mask, 32);
    }
    const float muf  = s * (1.0f / B_);
    const float var  = fmaxf(q * (1.0f / B_) - muf * muf, 0.0f);
    const float invf = 1.0f / (sqrtf(var) + EPS_);

    // ---- Preload per-feature weights, laid out for the WMMA lanes ----
    // Both A and B operands of 16x16x4 use: VGPR j, lane-half g -> kappa = j + 2g.
    // Chunk c covers h1 indices k = 4c .. 4c+3; this lane owns k = 4c + j + 2g.
    float w1p[5][2], b1p[5][2];
    v2f   w2a[5];                         // A operand: W2^T, row = n (j), K = kk
    #pragma unroll
    for (int c = 0; c < 5; ++c) {
        #pragma unroll
        for (int j = 0; j < 2; ++j) {
            const int kk  = 4 * c + j + 2 * g;
            const bool ki = (kk < H1_);
            w1p[c][j] = ki ? W1[(size_t)f * H1_ + kk] : 0.0f;
            b1p[c][j] = ki ? b1[(size_t)f * H1_ + kk] : 0.0f;
            w2a[c][j] = (ki && n < H2_) ? W2[(size_t)f * (H1_ * H2_) + kk * H2_ + n]
                                        : 0.0f;
        }
    }
    // D rows j = r + 8g live in register index r -> preload b2/W3 per row.
    float b2r[8], w3r[8];
    #pragma unroll
    for (int r = 0; r < 8; ++r) {
        const int j = r + 8 * g;
        b2r[r] = (j < H2_) ? b2[(size_t)f * H2_ + j] : 0.0f;
        w3r[r] = (j < H2_) ? W3[(size_t)f * H2_ + j] : 0.0f;
    }
    const float b3f = b3[f];

    // ---- Pass 2: 64 pairs of 16-sample tiles (two independent WMMA chains) ----
    float xv0 = x[(size_t)(0 + n) * F_ + f];
    float xv1 = x[(size_t)(16 + n) * F_ + f];
    for (int t = 0; t < (B_ / 32); ++t) {
        const int b0  = t * 32;
        const float xs0 = (xv0 - muf) * invf;
        const float xs1 = (xv1 - muf) * invf;

        // Prefetch next pair (clamped to a valid address on the last trip)
        const int bp = (t < (B_ / 32) - 1) ? (b0 + 32) : 0;
        xv0 = x[(size_t)(bp + 0 + n) * F_ + f];
        xv1 = x[(size_t)(bp + 16 + n) * F_ + f];

        v8f acc0 = {}, acc1 = {};
        #pragma unroll
        for (int c = 0; c < 5; ++c) {
            // Layer 1: h1[m][k] = relu(xs_m * W1[k] + b1[k]) -> B operand (h1^T)
            v2f hb0, hb1;
            hb0.x = fmaxf(xs0 * w1p[c][0] + b1p[c][0], 0.0f);
            hb0.y = fmaxf(xs0 * w1p[c][1] + b1p[c][1], 0.0f);
            hb1.x = fmaxf(xs1 * w1p[c][0] + b1p[c][0], 0.0f);
            hb1.y = fmaxf(xs1 * w1p[c][1] + b1p[c][1], 0.0f);
            // D[j,m] += sum_k W2[k,j] * h1[m,k]   (fp32 WMMA, A = W2^T chunk)
            acc0 = __builtin_amdgcn_wmma_f32_16x16x4_f32(
                false, w2a[c], false, hb0, (short)0, acc0, false, false);
            acc1 = __builtin_amdgcn_wmma_f32_16x16x4_f32(
                false, w2a[c], false, hb1, (short)0, acc1, false, false);
        }

        // Layers 2+3: register-local over j = r + 8g, then one cross-half combine
        float p0 = 0.f, p1 = 0.f;
        #pragma unroll
        for (int r = 0; r < 8; ++r) {
            p0 += fmaxf(acc0[r] + b2r[r], 0.0f) * w3r[r];
            p1 += fmaxf(acc1[r] + b2r[r], 0.0f) * w3r[r];
        }
        p0 += __shfl_xor(p0, 16, 32);   // add j=8..15 half to j=0..7 half
        p1 += __shfl_xor(p1, 16, 32);

        // Full-wave store: lanes 0-15 -> tile0 rows, lanes 16-31 -> tile1 rows
        const float oval = ((g == 0) ? p0 : p1) + b3f;
        out[(size_t)(b0 + lane) * F_ + f] = oval;   // sample b0 + lane
    }
}

extern "C" void kernel_launch(void* const* d_in, const int* in_sizes, int n_in,
                              void* d_out, int out_size, void* d_ws, size_t ws_size,
                              hipStream_t stream) {
    (void)in_sizes; (void)n_in; (void)out_size; (void)d_ws; (void)ws_size;
    const float* x  = (const float*)d_in[0];
    const float* W1 = (const float*)d_in[1];
    const float* b1 = (const float*)d_in[2];
    const float* W2 = (const float*)d_in[3];
    const float* b2 = (const float*)d_in[4];
    const float* W3 = (const float*)d_in[5];
    const float* b3 = (const float*)d_in[6];
    float* out = (float*)d_out;

    // 2048 features -> 2048 waves -> 256 blocks of 8 wave32's
    nam_mlp_wmma_kernel<<<dim3(F_ / 8), dim3(256), 0, stream>>>(
        x, W1, b1, W2, b2, W3, b3, out);
}